// SequentialPredictionLoss_51273319580286
// MI455X (gfx1250) — compile-verified
//
#include <hip/hip_runtime.h>
#include <hip/hip_bf16.h>

typedef __attribute__((ext_vector_type(2))) float v2f;
typedef __attribute__((ext_vector_type(4))) float v4f;
typedef __attribute__((ext_vector_type(8))) float v8f;

#define SMOOTH_THRESHOLD 0.2f
#define SMOOTH_WEIGHT    0.2f
#define POS_WEIGHT       2.0f
#define T_STEPS          16

__device__ __forceinline__ float pen1(float d) {
  float a = fmaxf(fabsf(d) - SMOOTH_THRESHOLD, 0.0f);
  return a * a;
}

// One block per batch element b. 256 threads = 8 waves; each wave owns R/8
// contiguous region columns. Streams region_scores once with NT b128 loads,
// accumulates pen per (t, lane) in f32, then reduces across the 32 lanes with
// V_WMMA_F32_16X16X4_F32 (A = acc matrix, B = ones) for exact f32 row sums.
__global__ __launch_bounds__(256) void smooth_partial_kernel(
    const float* __restrict__ rs, float* __restrict__ pen_sum, int R) {
  const int b    = blockIdx.x;
  const int tid  = threadIdx.x;
  const int wave = tid >> 5;
  const int lane = tid & 31;

  __shared__ float ldsT[8][16][33];   // per-wave A matrix: [t][source lane]
  __shared__ float wtot[8][15];       // per-wave row totals

  const float* base = rs + (size_t)b * T_STEPS * R;

  float acc[15];
#pragma unroll
  for (int t = 0; t < 15; ++t) acc[t] = 0.0f;

  const int r_per_wave = R >> 3;       // 256 for R=2048
  const int iters      = r_per_wave >> 7;  // 128 r per wave-iteration (4/lane)
  for (int it = 0; it < iters; ++it) {
    const int r0 = wave * r_per_wave + it * 128 + lane * 4;
    v4f prev = __builtin_nontemporal_load((const v4f*)(base + r0));
#pragma unroll
    for (int t = 1; t < T_STEPS; ++t) {
      v4f cur = __builtin_nontemporal_load((const v4f*)(base + (size_t)t * R + r0));
      acc[t - 1] += pen1(cur.x - prev.x) + pen1(cur.y - prev.y) +
                    pen1(cur.z - prev.z) + pen1(cur.w - prev.w);
      prev = cur;
    }
  }

  // Stage A[t][lane] (pad row t=15 with zeros).
#pragma unroll
  for (int t = 0; t < 15; ++t) ldsT[wave][t][lane] = acc[t];
  ldsT[wave][15][lane] = 0.0f;
  __syncthreads();

  // 8 chained WMMAs sweep K over the 32 source lanes, 4 at a time.
  // A 16x4 f32 fragment layout: lanes 0-15 -> (M=lane, K=0/1), lanes 16-31 -> (M=lane-16, K=2/3).
  const int m     = lane & 15;
  const int khalf = lane >> 4;
  v2f bones; bones.x = 1.0f; bones.y = 1.0f;
  v8f c = {};
#pragma unroll
  for (int i = 0; i < 8; ++i) {
    v2f a;
    a.x = ldsT[wave][m][4 * i + 2 * khalf + 0];
    a.y = ldsT[wave][m][4 * i + 2 * khalf + 1];
    c = __builtin_amdgcn_wmma_f32_16x16x4_f32(false, a, false, bones,
                                              (short)0, c, false, false);
  }

  // D[m][n] identical across n (B=ones). Lane 0 holds t=0..7 in c[0..7],
  // lane 16 holds t=8..15 (t=15 is the zero pad row).
  if (lane == 0) {
#pragma unroll
    for (int j = 0; j < 8; ++j) wtot[wave][j] = c[j];
  }
  if (lane == 16) {
#pragma unroll
    for (int j = 0; j < 7; ++j) wtot[wave][8 + j] = c[j];
  }
  __syncthreads();

  if (tid < 15) {
    float s = 0.0f;
#pragma unroll
    for (int w = 0; w < 8; ++w) s += wtot[w][tid];
    pen_sum[b * 15 + tid] = s;   // plain store; block b owns this row
  }
}

// Single-block finalize: masked BCE-with-logits + per-step smoothness logic.
__global__ __launch_bounds__(256) void finalize_kernel(
    const float* __restrict__ pred, const int* __restrict__ tgt,
    const int* __restrict__ vmask, const float* __restrict__ pen_sum,
    float* __restrict__ out, int B, int R) {
  const int tid = threadIdx.x;
  const int N = B * T_STEPS;

  float bce_sum = 0.0f, vm_sum = 0.0f;
  for (int i = tid; i < N; i += 256) {
    float vm = (vmask[i] != 0) ? 1.0f : 0.0f;
    float y  = (float)tgt[i];
    float x  = pred[i];
    float l1p    = log1pf(expf(-fabsf(x)));
    float ls_pos = fminf(x, 0.0f) - l1p;    // log_sigmoid(x)
    float ls_neg = fminf(-x, 0.0f) - l1p;   // log_sigmoid(-x)
    float bce = -(POS_WEIGHT * y * ls_pos + (1.0f - y) * ls_neg);
    bce_sum += bce * vm;
    vm_sum  += vm;
  }

  __shared__ float sred[256];
  sred[tid] = bce_sum;
  __syncthreads();
  for (int s = 128; s > 0; s >>= 1) {
    if (tid < s) sred[tid] += sred[tid + s];
    __syncthreads();
  }
  float bce_tot = sred[0];
  __syncthreads();
  sred[tid] = vm_sum;
  __syncthreads();
  for (int s = 128; s > 0; s >>= 1) {
    if (tid < s) sred[tid] += sred[tid + s];
    __syncthreads();
  }
  float vm_tot = sred[0];
  __syncthreads();

  __shared__ float s_cnt[15];
  __shared__ float s_sp[15];
  if (tid < 15) { s_cnt[tid] = 0.0f; s_sp[tid] = 0.0f; }
  __syncthreads();

  float lc[15], lsp[15];
#pragma unroll
  for (int t = 0; t < 15; ++t) { lc[t] = 0.0f; lsp[t] = 0.0f; }
  for (int b = tid; b < B; b += 256) {
    const int*   vmb = vmask + b * T_STEPS;
    const float* psb = pen_sum + b * 15;
#pragma unroll
    for (int t = 0; t < 15; ++t) {
      if ((vmb[t] != 0) && (vmb[t + 1] != 0)) {
        lc[t]  += 1.0f;
        lsp[t] += psb[t];
      }
    }
  }
#pragma unroll
  for (int t = 0; t < 15; ++t) {
    atomicAdd(&s_cnt[t], lc[t]);
    atomicAdd(&s_sp[t], lsp[t]);
  }
  __syncthreads();

  if (tid == 0) {
    float step_total = 0.0f;
    int   num_steps  = 0;
    const float invR = 1.0f / (float)R;
    for (int t = 0; t < 15; ++t) {
      float cnt = s_cnt[t];
      if (cnt > 0.0f) {
        step_total += (s_sp[t] * invR) / cnt;  // mean over R, then over valid b
        num_steps++;
      }
    }
    float smooth     = (num_steps > 0) ? (step_total / (float)num_steps) : 0.0f;
    float class_loss = bce_tot / fmaxf(vm_tot, 1e-8f);
    out[0] = class_loss + SMOOTH_WEIGHT * smooth;
  }
}

extern "C" void kernel_launch(void* const* d_in, const int* in_sizes, int n_in,
                              void* d_out, int out_size, void* d_ws, size_t ws_size,
                              hipStream_t stream) {
  const float* pred  = (const float*)d_in[0];   // [B, T, 1] f32
  const float* rs    = (const float*)d_in[1];   // [B, T, R, 1] f32
  const int*   tgt   = (const int*)d_in[2];     // [B, T] i32
  const int*   vmask = (const int*)d_in[3];     // [B, T] i32
  float* out     = (float*)d_out;
  float* pen_sum = (float*)d_ws;                // [B, 15] f32 scratch (fully overwritten)

  const int BT = in_sizes[0];        // B*T = 32768
  const int B  = BT / T_STEPS;       // 2048
  const int R  = in_sizes[1] / BT;   // 2048

  smooth_partial_kernel<<<B, 256, 0, stream>>>(rs, pen_sum, R);
  finalize_kernel<<<1, 256, 0, stream>>>(pred, tgt, vmask, pen_sum, out, B, R);
}